// SplitSignal_cplx_41798621724893
// MI455X (gfx1250) — compile-verified
//
#include <hip/hip_runtime.h>
#include <hip/hip_bf16.h>
#include <cstdint>

#define WINDOW   1024
#define HOP      256
#define T_LEN    262144
#define N_FRAMES 1021            // (262144 - 1024)/256 + 1
#define TILE     64
#define LDS_STRIDE 65            // 64 + 1 pad -> conflict-free transposed access

// out[ch, w, f] = X[ch, f*HOP + w]
// Tile the (f, w) plane 64x64. Phase 1: async global->LDS with transposing
// scatter (CDNA5 GLOBAL_LOAD_ASYNC_TO_LDS, tracked by ASYNCcnt). Phase 2:
// conflict-free LDS reads + fully coalesced non-temporal stores along f.
__global__ __launch_bounds__(256) void split_signal_kernel(const float* __restrict__ X,
                                                           float* __restrict__ out) {
    __shared__ float tile_lds[TILE * LDS_STRIDE];

    const int tid = threadIdx.x;
    const int ch  = blockIdx.y;            // 0..31  (= batch*2 channels)
    const int fT  = blockIdx.x & 15;       // frame-tile index   (0..15)
    const int wT  = blockIdx.x >> 4;       // window-tile index  (0..15)
    const int f0  = fT * TILE;
    const int w0  = wT * TILE;

    const float* __restrict__ src = X   + (size_t)ch * T_LEN;
    float*       __restrict__ dst = out + (size_t)ch * (size_t)WINDOW * N_FRAMES;

    // ---------------- Phase 1: async global -> LDS (transposing scatter) ----
    // Lanes along w => 128B-coalesced global reads per wave.
    // LDS dest = lds[w][f], stride 65 => banks (w+f)%64, conflict-free.
    const int wl = tid & 63;               // w within tile
    const int fr = tid >> 6;               // 0..3
    #pragma unroll
    for (int s = 0; s < 16; ++s) {
        const int f    = fr + 4 * s;       // 0..63 (lds slot, never clamped)
        int       fidx = f0 + f;
        if (fidx > N_FRAMES - 1) fidx = N_FRAMES - 1;   // keep reads in-bounds; value unused
        const float* gp = src + (size_t)fidx * HOP + (w0 + wl);
        const unsigned lds_off =
            (unsigned)(uintptr_t)(&tile_lds[wl * LDS_STRIDE + f]);
        asm volatile("global_load_async_to_lds_b32 %0, %1, off"
                     :: "v"(lds_off), "v"(gp)
                     : "memory");
    }
    asm volatile("s_wait_asynccnt 0" ::: "memory");
    __syncthreads();

    // ---------------- Phase 2: LDS -> global, coalesced along frames --------
    // Lanes along f => 128B-coalesced NT stores; LDS read banks (w+fx)%64.
    const int fx = tid & 63;
    const int wq = tid >> 6;               // 0..3
    const int f  = f0 + fx;
    if (f < N_FRAMES) {
        #pragma unroll
        for (int s = 0; s < 16; ++s) {
            const int w = wq + 4 * s;      // 0..63
            const float v = tile_lds[w * LDS_STRIDE + fx];
            __builtin_nontemporal_store(v, &dst[(size_t)(w0 + w) * N_FRAMES + f]);
        }
    }
}

extern "C" void kernel_launch(void* const* d_in, const int* in_sizes, int n_in,
                              void* d_out, int out_size, void* d_ws, size_t ws_size,
                              hipStream_t stream) {
    const float* X  = (const float*)d_in[0];
    float*      out = (float*)d_out;

    // 16 f-tiles * 16 w-tiles in x, 32 channels in y; 256 threads (8 waves).
    dim3 grid(256, 32, 1);
    dim3 block(256, 1, 1);
    split_signal_kernel<<<grid, block, 0, stream>>>(X, out);
}